// WindowAttention_292057776640
// MI455X (gfx1250) — compile-verified
//
#include <hip/hip_runtime.h>
#include <hip/hip_bf16.h>

typedef __attribute__((ext_vector_type(2))) float v2f;
typedef __attribute__((ext_vector_type(8))) float v8f;

#define NTOK  343
#define HEADS 12
#define HD    32
#define DIM   384
#define BATCH 128

union V8 { v8f v; float f[8]; };

static __device__ inline v8f wmma4(v2f a, v2f b, v8f c) {
  // D = A(16x4 f32) * B(4x16 f32) + C(16x16 f32), full fp32 matrix pipe
  return __builtin_amdgcn_wmma_f32_16x16x4_f32(false, a, false, b, (short)0, c,
                                               false, false);
}

// ---------------------------------------------------------------------------
// Kernel 1: QKV projection. C(43904,1152) = X(43904,384) @ W^T + b
// B tile is k-pair interleaved in LDS so each WMMA B-fragment is one b64 load.
// Epilogue scatters into q/k/v laid out (b, h, n, 32); q is pre-scaled.
// ---------------------------------------------------------------------------
__global__ __launch_bounds__(256) void qkv_kernel(
    const float* __restrict__ X, const float* __restrict__ W,
    const float* __restrict__ bias, float* __restrict__ Qo,
    float* __restrict__ Ko, float* __restrict__ Vo) {
  __shared__ float As[64][36];    // stride 36: aligned + conflict-free v2f
  __shared__ float Bsp[16][130];  // [k>>1][n*2+(k&1)], stride even

  const int t = threadIdx.x;
  const int lane = t & 31, wv = t >> 5;
  const int wm = wv >> 1, wn = wv & 1;
  const int ln = lane & 15, hi = lane >> 4;
  const int m0 = blockIdx.x * 64, n0 = blockIdx.y * 64;

  V8 acc0, acc1;
#pragma unroll
  for (int r = 0; r < 8; ++r) { acc0.f[r] = 0.f; acc1.f[r] = 0.f; }

  const int am = t >> 2, akq = (t & 3) << 3;  // 64 rows x 32 k, 8 k/thread
  const int bn = t >> 2, bkq = (t & 3) << 3;

  for (int kk = 0; kk < DIM; kk += 32) {
    float4 a0 = *(const float4*)&X[(m0 + am) * DIM + kk + akq];
    float4 a1 = *(const float4*)&X[(m0 + am) * DIM + kk + akq + 4];
    float4 b0 = *(const float4*)&W[(n0 + bn) * DIM + kk + bkq];
    float4 b1 = *(const float4*)&W[(n0 + bn) * DIM + kk + bkq + 4];
    if (kk + 32 < DIM) {
      __builtin_prefetch(&X[(m0 + am) * DIM + kk + 32 + akq], 0, 1);
      __builtin_prefetch(&W[(n0 + bn) * DIM + kk + 32 + bkq], 0, 1);
    }
    *(float4*)&As[am][akq]     = a0;
    *(float4*)&As[am][akq + 4] = a1;
    const int bk2 = bkq >> 1;  // 0,4,8,12
    *(float2*)&Bsp[bk2 + 0][bn * 2] = make_float2(b0.x, b0.y);
    *(float2*)&Bsp[bk2 + 1][bn * 2] = make_float2(b0.z, b0.w);
    *(float2*)&Bsp[bk2 + 2][bn * 2] = make_float2(b1.x, b1.y);
    *(float2*)&Bsp[bk2 + 3][bn * 2] = make_float2(b1.z, b1.w);
    __syncthreads();
#pragma unroll
    for (int ks = 0; ks < 32; ks += 4) {
      const int k0 = ks + 2 * hi;
      v2f a  = *(const v2f*)&As[16 * wm + ln][k0];
      v2f bb0 = *(const v2f*)&Bsp[k0 >> 1][(32 * wn + ln) * 2];
      v2f bb1 = *(const v2f*)&Bsp[k0 >> 1][(32 * wn + 16 + ln) * 2];
      acc0.v = wmma4(a, bb0, acc0.v);
      acc1.v = wmma4(a, bb1, acc1.v);
    }
    __syncthreads();
  }

#pragma unroll
  for (int j = 0; j < 2; ++j) {
    V8& acc = j ? acc1 : acc0;
    const int colbase = n0 + 32 * wn + 16 * j;  // multiple of 16
    const int s   = colbase / DIM;              // 0=q 1=k 2=v
    const int rem = colbase - s * DIM;
    const int h   = rem >> 5;
    const int ddb = rem & 31;
    const float bv  = bias[colbase + ln];
    float* dst      = (s == 0) ? Qo : (s == 1) ? Ko : Vo;
    const float scl = (s == 0) ? 0.17677669529663687f : 1.0f;  // 1/sqrt(32)
#pragma unroll
    for (int r = 0; r < 8; ++r) {
      const int grow = m0 + 16 * wm + 8 * hi + r;
      const int bb = grow / NTOK;
      const int nn = grow - bb * NTOK;
      dst[((bb * HEADS + h) * NTOK + nn) * HD + ddb + ln] = (acc.f[r] + bv) * scl;
    }
  }
}

// ---------------------------------------------------------------------------
// Kernel 2: fused attention per (b,h): S = qK^T + bias + mask; softmax; O=PV
// grid.x = B*HEADS, grid.y = 11 (q tiles of 32). S lives entirely in LDS.
// ---------------------------------------------------------------------------
__global__ __launch_bounds__(256) void attn_kernel(
    const float* __restrict__ Q, const float* __restrict__ K,
    const float* __restrict__ V, const float* __restrict__ mask,
    const float* __restrict__ bias_table, const int* __restrict__ rel_idx,
    float* __restrict__ O) {
  __shared__ float Sq[32][36];     // q tile, aligned/conflict-free v2f frags
  __shared__ float Stage[2112];    // phase1: Kt [16][130]; phase3: V [32][66]
  __shared__ float Sb[32][346];    // scores -> probabilities (even stride)

  const int t = threadIdx.x;
  const int lane = t & 31, wv = t >> 5;
  const int ln = lane & 15, hi = lane >> 4;
  const int bh = blockIdx.x;
  const int b = bh / HEADS, h = bh - b * HEADS;
  const int q0 = blockIdx.y * 32;
  const float* Qb = Q + (size_t)bh * NTOK * HD;
  const float* Kb = K + (size_t)bh * NTOK * HD;
  const float* Vb = V + (size_t)bh * NTOK * HD;
  const float* Mb = mask + (size_t)(b & 63) * NTOK * NTOK;

  {  // load q tile (rows past 343 -> zero)
    const int qr = t >> 3, dq = (t & 7) * 4;
    float4 qv = make_float4(0.f, 0.f, 0.f, 0.f);
    if (q0 + qr < NTOK) qv = *(const float4*)&Qb[(q0 + qr) * HD + dq];
    *(float4*)&Sq[qr][dq] = qv;
  }

  // ---- Phase 1: scores -------------------------------------------------
  const int qh1 = wv >> 2, kt4 = wv & 3;  // 8 waves cover 32q x 64k
  for (int c = 0; c < 6; ++c) {
    __syncthreads();
    {  // stage K chunk pair-interleaved transposed: [dd>>1][key*2+(dd&1)]
      const int key = t >> 2, dq = (t & 3) * 8;
      const int kg = c * 64 + key;
      float4 v0 = make_float4(0.f, 0.f, 0.f, 0.f), v1 = v0;
      if (kg < NTOK) {
        v0 = *(const float4*)&Kb[kg * HD + dq];
        v1 = *(const float4*)&Kb[kg * HD + dq + 4];
      }
      const int r2 = dq >> 1;  // 0,4,8,12
      *(float2*)&Stage[(r2 + 0) * 130 + key * 2] = make_float2(v0.x, v0.y);
      *(float2*)&Stage[(r2 + 1) * 130 + key * 2] = make_float2(v0.z, v0.w);
      *(float2*)&Stage[(r2 + 2) * 130 + key * 2] = make_float2(v1.x, v1.y);
      *(float2*)&Stage[(r2 + 3) * 130 + key * 2] = make_float2(v1.z, v1.w);
    }
    __syncthreads();
    V8 acc;
#pragma unroll
    for (int r = 0; r < 8; ++r) acc.f[r] = 0.f;
#pragma unroll
    for (int ks = 0; ks < 32; ks += 4) {
      const int k0 = ks + 2 * hi;
      v2f a  = *(const v2f*)&Sq[16 * qh1 + ln][k0];
      v2f bb = *(const v2f*)&Stage[(k0 >> 1) * 130 + (16 * kt4 + ln) * 2];
      acc.v = wmma4(a, bb, acc.v);
    }
#pragma unroll
    for (int r = 0; r < 8; ++r) {
      const int qloc = 16 * qh1 + 8 * hi + r;
      const int qg = q0 + qloc;
      const int kg = c * 64 + 16 * kt4 + ln;
      if (kg < 344) {
        float val = -1e30f;
        if (kg < NTOK && qg < NTOK) {
          const int ri = rel_idx[qg * NTOK + kg];
          val = acc.f[r] + bias_table[ri * HEADS + h] + Mb[qg * NTOK + kg];
        }
        Sb[qloc][kg] = val;
      }
    }
  }
  __syncthreads();

  // ---- Phase 2: softmax (8 lanes per row) ------------------------------
  {
    const int row = t >> 3, l8 = t & 7;
    float mx = -1e30f;
    for (int j = l8; j < 344; j += 8) mx = fmaxf(mx, Sb[row][j]);
    mx = fmaxf(mx, __shfl_xor(mx, 1, 32));
    mx = fmaxf(mx, __shfl_xor(mx, 2, 32));
    mx = fmaxf(mx, __shfl_xor(mx, 4, 32));
    float sum = 0.f;
    for (int j = l8; j < 344; j += 8) {
      const float e = expf(Sb[row][j] - mx);
      Sb[row][j] = e;
      sum += e;
    }
    sum += __shfl_xor(sum, 1, 32);
    sum += __shfl_xor(sum, 2, 32);
    sum += __shfl_xor(sum, 4, 32);
    const float inv = 1.f / sum;
    for (int j = l8; j < 344; j += 8) Sb[row][j] *= inv;
  }

  // ---- Phase 3: O = P @ V ----------------------------------------------
  V8 acc;
#pragma unroll
  for (int r = 0; r < 8; ++r) acc.f[r] = 0.f;
  const int qh3 = wv >> 1, dh = wv & 1;  // waves 0..3 own the 4 O tiles
  for (int c = 0; c < 6; ++c) {
    __syncthreads();
    {  // stage V chunk pair-interleaved over key: [key>>1][dd*2+(key&1)]
      const int key = t >> 2, dq = (t & 3) * 8;
      const int kg = c * 64 + key;
      float4 v0 = make_float4(0.f, 0.f, 0.f, 0.f), v1 = v0;
      if (kg < NTOK) {
        v0 = *(const float4*)&Vb[kg * HD + dq];
        v1 = *(const float4*)&Vb[kg * HD + dq + 4];
      }
      float* s = &Stage[(key >> 1) * 66 + (key & 1)];
      s[(dq + 0) * 2] = v0.x; s[(dq + 1) * 2] = v0.y;
      s[(dq + 2) * 2] = v0.z; s[(dq + 3) * 2] = v0.w;
      s[(dq + 4) * 2] = v1.x; s[(dq + 5) * 2] = v1.y;
      s[(dq + 6) * 2] = v1.z; s[(dq + 7) * 2] = v1.w;
    }
    __syncthreads();
    if (wv < 4) {
      if (c < 5) {
#pragma unroll
        for (int ks = 0; ks < 64; ks += 4) {
          const int k0 = ks + 2 * hi;
          const int kg = c * 64 + k0;
          v2f a  = *(const v2f*)&Sb[16 * qh3 + ln][kg];
          v2f bb = *(const v2f*)&Stage[(k0 >> 1) * 66 + (16 * dh + ln) * 2];
          acc.v = wmma4(a, bb, acc.v);
        }
      } else {
#pragma unroll
        for (int ks = 0; ks < 64; ks += 4) {
          const int k0 = ks + 2 * hi;
          const int kg = 5 * 64 + k0;
          v2f a;
          a.x = Sb[16 * qh3 + ln][min(kg, 343)];      // P[.,343]==0 exactly
          a.y = Sb[16 * qh3 + ln][min(kg + 1, 343)];
          v2f bb = *(const v2f*)&Stage[(k0 >> 1) * 66 + (16 * dh + ln) * 2];
          acc.v = wmma4(a, bb, acc.v);
        }
      }
    }
  }
  if (wv < 4) {
#pragma unroll
    for (int r = 0; r < 8; ++r) {
      const int qg = q0 + 16 * qh3 + 8 * hi + r;
      if (qg < NTOK)
        O[((size_t)b * NTOK + qg) * DIM + h * HD + 16 * dh + ln] = acc.f[r];
    }
  }
}

// ---------------------------------------------------------------------------
// Kernel 3: output projection. out(43904,384) = att @ w_proj^T + b_proj
// ---------------------------------------------------------------------------
__global__ __launch_bounds__(256) void proj_kernel(
    const float* __restrict__ A, const float* __restrict__ W,
    const float* __restrict__ bias, float* __restrict__ C) {
  __shared__ float As[64][36];
  __shared__ float Bsp[16][130];

  const int t = threadIdx.x;
  const int lane = t & 31, wv = t >> 5;
  const int wm = wv >> 1, wn = wv & 1;
  const int ln = lane & 15, hi = lane >> 4;
  const int m0 = blockIdx.x * 64, n0 = blockIdx.y * 64;

  V8 acc0, acc1;
#pragma unroll
  for (int r = 0; r < 8; ++r) { acc0.f[r] = 0.f; acc1.f[r] = 0.f; }

  const int am = t >> 2, akq = (t & 3) << 3;
  const int bn = t >> 2, bkq = (t & 3) << 3;

  for (int kk = 0; kk < DIM; kk += 32) {
    float4 a0 = *(const float4*)&A[(m0 + am) * DIM + kk + akq];
    float4 a1 = *(const float4*)&A[(m0 + am) * DIM + kk + akq + 4];
    float4 b0 = *(const float4*)&W[(n0 + bn) * DIM + kk + bkq];
    float4 b1 = *(const float4*)&W[(n0 + bn) * DIM + kk + bkq + 4];
    if (kk + 32 < DIM) {
      __builtin_prefetch(&A[(m0 + am) * DIM + kk + 32 + akq], 0, 1);
      __builtin_prefetch(&W[(n0 + bn) * DIM + kk + 32 + bkq], 0, 1);
    }
    *(float4*)&As[am][akq]     = a0;
    *(float4*)&As[am][akq + 4] = a1;
    const int bk2 = bkq >> 1;
    *(float2*)&Bsp[bk2 + 0][bn * 2] = make_float2(b0.x, b0.y);
    *(float2*)&Bsp[bk2 + 1][bn * 2] = make_float2(b0.z, b0.w);
    *(float2*)&Bsp[bk2 + 2][bn * 2] = make_float2(b1.x, b1.y);
    *(float2*)&Bsp[bk2 + 3][bn * 2] = make_float2(b1.z, b1.w);
    __syncthreads();
#pragma unroll
    for (int ks = 0; ks < 32; ks += 4) {
      const int k0 = ks + 2 * hi;
      v2f a   = *(const v2f*)&As[16 * wm + ln][k0];
      v2f bb0 = *(const v2f*)&Bsp[k0 >> 1][(32 * wn + ln) * 2];
      v2f bb1 = *(const v2f*)&Bsp[k0 >> 1][(32 * wn + 16 + ln) * 2];
      acc0.v = wmma4(a, bb0, acc0.v);
      acc1.v = wmma4(a, bb1, acc1.v);
    }
    __syncthreads();
  }

#pragma unroll
  for (int j = 0; j < 2; ++j) {
    V8& acc = j ? acc1 : acc0;
    const int col = n0 + 32 * wn + 16 * j + ln;
    const float bv = bias[col];
#pragma unroll
    for (int r = 0; r < 8; ++r)
      C[(m0 + 16 * wm + 8 * hi + r) * DIM + col] = acc.f[r] + bv;
  }
}

// ---------------------------------------------------------------------------
extern "C" void kernel_launch(void* const* d_in, const int* in_sizes, int n_in,
                              void* d_out, int out_size, void* d_ws,
                              size_t ws_size, hipStream_t stream) {
  (void)in_sizes; (void)n_in; (void)out_size; (void)ws_size;
  const float* x          = (const float*)d_in[0];
  const float* mask       = (const float*)d_in[1];
  const float* w_qkv      = (const float*)d_in[2];
  const float* b_qkv      = (const float*)d_in[3];
  const float* bias_table = (const float*)d_in[4];
  const float* w_proj     = (const float*)d_in[5];
  const float* b_proj     = (const float*)d_in[6];
  const int*   rel_idx    = (const int*)d_in[7];

  const size_t per = (size_t)BATCH * HEADS * NTOK * HD;  // 16,859,136 floats
  float* qb  = (float*)d_ws;
  float* kb  = qb + per;
  float* vb  = kb + per;
  float* att = vb + per;

  // M = 128*343 = 43904 = 686*64 exactly
  qkv_kernel<<<dim3(686, 18), 256, 0, stream>>>(x, w_qkv, b_qkv, qb, kb, vb);
  attn_kernel<<<dim3(BATCH * HEADS, 11), 256, 0, stream>>>(
      qb, kb, vb, mask, bias_table, rel_idx, att);
  proj_kernel<<<dim3(686, 6), 256, 0, stream>>>(att, w_proj, b_proj,
                                                (float*)d_out);
}